// KernelPool_64811056497337
// MI455X (gfx1250) — compile-verified
//
#include <hip/hip_runtime.h>
#include <hip/hip_bf16.h>

// Problem constants (from reference)
#define NBC    128          // 8*16 batch*channel pairs
#define N_IN   4096
#define N_OUT  512
#define GAMMA  0.5f         // 0.5 / sigma^2, sigma = 1
#define ALPHA  1e-6f

typedef __attribute__((ext_vector_type(2))) float v2f;
typedef __attribute__((ext_vector_type(8))) float v8f;

__device__ __forceinline__ v8f wmma_f32_16x16x4(v2f a, v2f b, v8f c)
{
    return __builtin_amdgcn_wmma_f32_16x16x4_f32(
        /*neg_a=*/false, a, /*neg_b=*/false, b,
        /*c_mod=*/(short)0, c, /*reuse_a=*/false, /*reuse_b=*/false);
}

// ---------------------------------------------------------------------------
// Kernel 1: top-512 by |w| (descending, ties -> lower index first, matching
// jax.lax.top_k) via full bitonic sort of packed 64-bit keys in LDS, then
// gather the selected positions directly into d_out.
// ---------------------------------------------------------------------------
__global__ __launch_bounds__(512) void topk_gather(
    const float* __restrict__ in_pos,   // (128, 4096, 3)
    const float* __restrict__ in_w,     // (128, 4096)
    float* __restrict__ out_pos)        // (128, 512, 3)  [d_out region 0]
{
    __shared__ unsigned long long keys[N_IN];
    const int bc  = blockIdx.x;
    const int tid = threadIdx.x;
    const float* wbc = in_w + (size_t)bc * N_IN;

    #pragma unroll
    for (int e = 0; e < N_IN / 512; ++e) {
        int i = e * 512 + tid;
        unsigned int bits = __float_as_uint(fabsf(wbc[i]));  // |w| >= 0 -> monotonic bits
        keys[i] = ((unsigned long long)bits << 12) |
                  (unsigned long long)(N_IN - 1 - i);        // tie-break: small idx wins
    }
    __syncthreads();

    // Bitonic sort, descending on packed key.
    for (int k = 2; k <= N_IN; k <<= 1) {
        for (int j = k >> 1; j > 0; j >>= 1) {
            #pragma unroll
            for (int e = 0; e < N_IN / 512; ++e) {
                int i   = e * 512 + tid;
                int ixj = i ^ j;
                if (ixj > i) {
                    unsigned long long a = keys[i], b = keys[ixj];
                    bool desc = ((i & k) == 0);
                    bool sw   = desc ? (a < b) : (a > b);
                    if (sw) { keys[i] = b; keys[ixj] = a; }
                }
            }
            __syncthreads();
        }
    }

    // First 512 entries are the top-k in order.
    unsigned long long p = keys[tid];
    int src = N_IN - 1 - (int)(p & 0xFFFULL);
    const float* sp = in_pos + ((size_t)bc * N_IN + src) * 3;
    float*       dp = out_pos + ((size_t)bc * N_OUT + tid) * 3;
    dp[0] = sp[0]; dp[1] = sp[1]; dp[2] = sp[2];
}

// ---------------------------------------------------------------------------
// WMMA operand builders for the folded Gaussian-kernel Gram:
//   D = A(16x4) * B(4x16) + C  with
//   A row M = (x, y, z, 1)                      [lanes 0-15: K0,K1 ; 16-31: K2,K3]
//   B col N = (-2g x, -2g y, -2g z, -g*|p|^2)   [lanes 0-15: K0,K1 ; 16-31: K2,K3]
//   C       = -g*|o|^2 per row                  [VGPR r -> M = r + 8*(lane>=16)]
// giving D = -gamma * ||o - i||^2 per element.
// ---------------------------------------------------------------------------
__device__ __forceinline__ v2f make_a(const float* X, const float* Y,
                                      const float* Z, int base, int lane)
{
    int m = base + (lane & 15);
    float x = X[m], y = Y[m], z = Z[m];
    bool hi = lane >= 16;
    v2f a;
    a.x = hi ? z : x;
    a.y = hi ? 1.0f : y;
    return a;
}

__device__ __forceinline__ v2f make_b(const float* X, const float* Y,
                                      const float* Z, const float* Ng, // -g*|p|^2
                                      int base, int lane)
{
    int n = base + (lane & 15);
    float x = -2.0f * GAMMA * X[n];
    float y = -2.0f * GAMMA * Y[n];
    float z = -2.0f * GAMMA * Z[n];
    float q = Ng[n];
    bool hi = lane >= 16;
    v2f b;
    b.x = hi ? z : x;
    b.y = hi ? q : y;
    return b;
}

// ---------------------------------------------------------------------------
// Kernel 2: build samples = K_oi @ w and K_oo + alpha*I (to scratch).
// One block per (b,c), 8 waves. Inputs staged through LDS in 1024 chunks,
// with a prefetch of the next chunk (global_prefetch_b8).
// ---------------------------------------------------------------------------
#define CHUNK 1024

__global__ __launch_bounds__(256) void build_kernels(
    const float* __restrict__ in_pos,
    const float* __restrict__ in_w,
    const float* __restrict__ out_pos,   // d_out region 0 (written by kernel 1)
    float* __restrict__ Kmat,            // ws: 128 * 512 * 512 f32
    float* __restrict__ samples)         // ws: 128 * 512 f32
{
    __shared__ float ox[N_OUT], oy[N_OUT], oz[N_OUT], on[N_OUT]; // on = -g*|o|^2
    __shared__ float cx[CHUNK], cy[CHUNK], cz[CHUNK], cn[CHUNK], cw[CHUNK];

    const int bc   = blockIdx.x;
    const int tid  = threadIdx.x;
    const int lane = tid & 31;
    const int wave = tid >> 5;

    // Stage output positions + folded norms.
    const float* op = out_pos + (size_t)bc * N_OUT * 3;
    for (int i = tid; i < N_OUT; i += 256) {
        float x = op[i * 3 + 0], y = op[i * 3 + 1], z = op[i * 3 + 2];
        ox[i] = x; oy[i] = y; oz[i] = z;
        on[i] = -GAMMA * (x * x + y * y + z * z);
    }
    __syncthreads();

    // ---- samples = K_oi @ w -------------------------------------------------
    // Row tiles: wave handles tiles (wave + 8*rt), rt = 0..3  (32 tiles total).
    float acc[4][8];
    #pragma unroll
    for (int rt = 0; rt < 4; ++rt)
        #pragma unroll
        for (int r = 0; r < 8; ++r) acc[rt][r] = 0.0f;

    const float* ip = in_pos + (size_t)bc * N_IN * 3;
    const float* iw = in_w + (size_t)bc * N_IN;

    for (int ch = 0; ch < N_IN / CHUNK; ++ch) {
        __syncthreads();
        for (int i = tid; i < CHUNK; i += 256) {
            int g = ch * CHUNK + i;
            float x = ip[g * 3 + 0], y = ip[g * 3 + 1], z = ip[g * 3 + 2];
            cx[i] = x; cy[i] = y; cz[i] = z;
            cn[i] = -GAMMA * (x * x + y * y + z * z);
            cw[i] = iw[g];
        }
        if (ch + 1 < N_IN / CHUNK) {                 // prefetch next chunk
            int g = (ch + 1) * CHUNK + tid;
            __builtin_prefetch(&ip[g * 3], 0, 3);
            __builtin_prefetch(&iw[g], 0, 3);
        }
        __syncthreads();

        #pragma unroll
        for (int rt = 0; rt < 4; ++rt) {
            int row0 = (wave + rt * 8) * 16;
            v2f a = make_a(ox, oy, oz, row0, lane);
            int mb = row0 + 8 * (lane >> 4);
            for (int ct = 0; ct < CHUNK / 16; ++ct) {
                int col0 = ct * 16;
                v2f b = make_b(cx, cy, cz, cn, col0, lane);
                v8f c;
                #pragma unroll
                for (int r = 0; r < 8; ++r) c[r] = on[mb + r];
                c = wmma_f32_16x16x4(a, b, c);          // c[r] = -g*||o-i||^2
                float wN = cw[col0 + (lane & 15)];
                #pragma unroll
                for (int r = 0; r < 8; ++r) {
                    float kv = __expf(fminf(c[r], 0.0f));
                    acc[rt][r] += kv * wN;
                }
            }
        }
    }
    __syncthreads();

    // Reduce each row's 16 lane-partials and write samples.
    #pragma unroll
    for (int rt = 0; rt < 4; ++rt) {
        #pragma unroll
        for (int r = 0; r < 8; ++r) {
            float v = acc[rt][r];
            v += __shfl_xor(v, 1, 16);
            v += __shfl_xor(v, 2, 16);
            v += __shfl_xor(v, 4, 16);
            v += __shfl_xor(v, 8, 16);
            if ((lane & 15) == 0) {
                int m = (wave + rt * 8) * 16 + r + 8 * (lane >> 4);
                samples[(size_t)bc * N_OUT + m] = v;
            }
        }
    }

    // ---- K_oo + alpha*I -> scratch -----------------------------------------
    float* Kbc = Kmat + (size_t)bc * N_OUT * N_OUT;
    for (int t = wave; t < 32 * 32; t += 8) {
        int row0 = (t >> 5) * 16;
        int col0 = (t & 31) * 16;
        v2f a = make_a(ox, oy, oz, row0, lane);
        v2f b = make_b(ox, oy, oz, on, col0, lane);
        int mb = row0 + 8 * (lane >> 4);
        v8f c;
        #pragma unroll
        for (int r = 0; r < 8; ++r) c[r] = on[mb + r];
        c = wmma_f32_16x16x4(a, b, c);
        int gn = col0 + (lane & 15);
        #pragma unroll
        for (int r = 0; r < 8; ++r) {
            int gm = mb + r;
            float kv = __expf(fminf(c[r], 0.0f));
            if (gm == gn) kv += ALPHA;
            Kbc[(size_t)gm * N_OUT + gn] = kv;
        }
    }
}

// ---------------------------------------------------------------------------
// Kernel 3: SPD solve per (b,c): blocked right-looking Cholesky with panel
// width 16. Panel factorization is column-parallel in LDS; the rank-16
// trailing update A_tile -= L_i * L_j^T runs on the matrix units as a chain
// of 4 x V_WMMA_F32_16X16X4_F32 (subtraction folded by negating B).
// Then forward/backward substitution. 16 waves; tile ids are wave-uniform
// so EXEC is all-ones at every WMMA.
// ---------------------------------------------------------------------------
#define NPANEL 16               // panel width
#define NBLK   (N_OUT / NPANEL) // 32 block rows/cols

__global__ __launch_bounds__(512) void cholesky_solve(
    float* __restrict__ Kmat,
    const float* __restrict__ samples,
    float* __restrict__ out_w)          // d_out region 1
{
    __shared__ float P[N_OUT * NPANEL]; // current panel L[:, k0:k0+16] (32 KB)
    __shared__ float bvec[N_OUT];
    __shared__ float xv[N_OUT];
    __shared__ float sc;

    const int bc   = blockIdx.x;
    const int tid  = threadIdx.x;
    const int lane = tid & 31;
    const int wave = tid >> 5;
    float* A = Kmat + (size_t)bc * N_OUT * N_OUT;

    for (int i = tid; i < N_OUT; i += 512)
        bvec[i] = samples[(size_t)bc * N_OUT + i];
    __syncthreads();

    for (int kb = 0; kb < NBLK; ++kb) {
        const int k0 = kb * NPANEL;

        // Stage panel A[k0:512, k0:k0+16] into LDS (row-major, 16 per row).
        for (int i = k0 + tid; i < N_OUT; i += 512) {
            #pragma unroll
            for (int c = 0; c < NPANEL; ++c)
                P[i * NPANEL + c] = A[(size_t)i * N_OUT + k0 + c];
        }
        __syncthreads();

        // Factor the panel (diagonal block + column scaling + rank-1 updates).
        for (int c = 0; c < NPANEL; ++c) {
            if (tid == 0) {
                float d = sqrtf(P[(k0 + c) * NPANEL + c]);
                P[(k0 + c) * NPANEL + c] = d;
                sc = 1.0f / d;
            }
            __syncthreads();
            float inv = sc;
            for (int i = k0 + c + 1 + tid; i < N_OUT; i += 512)
                P[i * NPANEL + c] *= inv;
            __syncthreads();
            for (int i = k0 + c + 1 + tid; i < N_OUT; i += 512) {
                float vic = P[i * NPANEL + c];
                #pragma unroll
                for (int j = c + 1; j < NPANEL; ++j)
                    P[i * NPANEL + j] -= vic * P[(k0 + j) * NPANEL + c];
            }
            __syncthreads();
        }

        // Write the factored panel back to global A.
        for (int i = k0 + tid; i < N_OUT; i += 512) {
            #pragma unroll
            for (int c = 0; c < NPANEL; ++c)
                A[(size_t)i * N_OUT + k0 + c] = P[i * NPANEL + c];
        }
        __syncthreads();

        // Trailing update on matrix units: tiles (ti, tj), kb < tj <= ti.
        const int nt    = NBLK - 1 - kb;        // block rows below the panel
        const int total = nt * (nt + 1) / 2;
        for (int t = wave; t < total; t += 16) {
            int r = (int)((sqrtf(8.0f * (float)t + 1.0f) - 1.0f) * 0.5f);
            while (((r + 1) * (r + 2)) / 2 <= t) ++r;
            while ((r * (r + 1)) / 2 > t) --r;
            const int ti  = kb + 1 + r;
            const int tj  = kb + 1 + (t - (r * (r + 1)) / 2);
            const int gm0 = ti * NPANEL;
            const int gn0 = tj * NPANEL;
            const int n   = lane & 15;
            const int mrw = 8 * (lane >> 4);
            const bool hi = lane >= 16;

            v8f cacc;
            #pragma unroll
            for (int r8 = 0; r8 < 8; ++r8)
                cacc[r8] = A[(size_t)(gm0 + mrw + r8) * N_OUT + gn0 + n];

            #pragma unroll
            for (int s = 0; s < 4; ++s) {       // K = 16 in 4 slices of 4
                const int k4 = s * 4;
                v2f a, b;
                a.x =  P[(gm0 + n) * NPANEL + k4 + (hi ? 2 : 0)];
                a.y =  P[(gm0 + n) * NPANEL + k4 + (hi ? 3 : 1)];
                b.x = -P[(gn0 + n) * NPANEL + k4 + (hi ? 2 : 0)];
                b.y = -P[(gn0 + n) * NPANEL + k4 + (hi ? 3 : 1)];
                cacc = wmma_f32_16x16x4(a, b, cacc);   // C - L_i(:,k4..) L_j(:,k4..)^T
            }

            #pragma unroll
            for (int r8 = 0; r8 < 8; ++r8)
                A[(size_t)(gm0 + mrw + r8) * N_OUT + gn0 + n] = cacc[r8];
        }
        __syncthreads();
    }

    // Forward solve L y = b (y overwrites bvec).
    for (int k = 0; k < N_OUT; ++k) {
        if (tid == 0) bvec[k] = bvec[k] / A[(size_t)k * N_OUT + k];
        __syncthreads();
        float yk = bvec[k];
        for (int i = k + 1 + tid; i < N_OUT; i += 512)
            bvec[i] -= A[(size_t)i * N_OUT + k] * yk;
        __syncthreads();
    }

    // Backward solve L^T x = y.
    for (int k = N_OUT - 1; k >= 0; --k) {
        if (tid == 0) {
            float v = bvec[k] / A[(size_t)k * N_OUT + k];
            xv[k] = v; bvec[k] = v;
        }
        __syncthreads();
        float xk = bvec[k];
        for (int i = tid; i < k; i += 512)
            bvec[i] -= A[(size_t)k * N_OUT + i] * xk;
        __syncthreads();
    }

    for (int i = tid; i < N_OUT; i += 512)
        out_w[(size_t)bc * N_OUT + i] = xv[i];
}

// ---------------------------------------------------------------------------
// Launch: d_in[0]=positions (8,16,4096,3) f32, d_in[1]=weights (8,16,4096) f32
// d_out = [output_positions (8,16,512,3) | output_weights (8,16,512)] f32
// d_ws  = [K matrices 128*512*512 f32 (128 MB) | samples 128*512 f32]
// ---------------------------------------------------------------------------
extern "C" void kernel_launch(void* const* d_in, const int* in_sizes, int n_in,
                              void* d_out, int out_size, void* d_ws, size_t ws_size,
                              hipStream_t stream)
{
    (void)in_sizes; (void)n_in; (void)out_size; (void)ws_size;

    const float* in_pos = (const float*)d_in[0];
    const float* in_w   = (const float*)d_in[1];

    float* out_pos = (float*)d_out;                                 // NBC*512*3
    float* out_w   = (float*)d_out + (size_t)NBC * N_OUT * 3;       // NBC*512

    float* Kmat    = (float*)d_ws;                                  // NBC*512*512
    float* samples = (float*)d_ws + (size_t)NBC * N_OUT * N_OUT;    // NBC*512

    topk_gather  <<<NBC, 512, 0, stream>>>(in_pos, in_w, out_pos);
    build_kernels<<<NBC, 256, 0, stream>>>(in_pos, in_w, out_pos, Kmat, samples);
    cholesky_solve<<<NBC, 512, 0, stream>>>(Kmat, samples, out_w);
}